// LightGCN_40544491274543
// MI455X (gfx1250) — compile-verified
//
#include <hip/hip_runtime.h>
#include <hip/hip_bf16.h>

// Problem constants (from the reference)
#define U_CNT 200000
#define I_CNT 100000
#define N_CNT (U_CNT + I_CNT)
#define DIMS  64
#define E_CNT 9600000

static constexpr long UD = (long)U_CNT * DIMS;   // 12,800,000
static constexpr long ID = (long)I_CNT * DIMS;   //  6,400,000
static constexpr long ND = (long)N_CNT * DIMS;   // 19,200,000

// ---------------------------------------------------------------------------
// init: x = csum = concat(user_emb, item_emb); y = 0   (all float4 / b128)
// ---------------------------------------------------------------------------
__global__ void lightgcn_init(const float* __restrict__ ue,
                              const float* __restrict__ ie,
                              float* __restrict__ x,
                              float* __restrict__ csum,
                              float* __restrict__ y) {
  long i4 = (long)blockIdx.x * blockDim.x + threadIdx.x;   // float4 index
  const long n4 = ND / 4;
  if (i4 >= n4) return;
  const long u4 = UD / 4;
  float4 v;
  if (i4 < u4) v = ((const float4*)ue)[i4];
  else         v = ((const float4*)ie)[i4 - u4];
  ((float4*)x)[i4]    = v;
  ((float4*)csum)[i4] = v;
  ((float4*)y)[i4]    = make_float4(0.f, 0.f, 0.f, 0.f);
}

// ---------------------------------------------------------------------------
// Async edge staging: 32 edges' (row, col, vals) -> LDS via the ASYNCcnt path.
// Each lane writes one 16B interleaved record {row, col, valbits, pad} so the
// consumer loop needs a single ds_load_b128 broadcast per edge.
// LDS address = low 32 bits of the generic shared pointer (aperture lives in
// the high 32 bits; HW adds the wave's LDS_BASE per ISA 15.18).
// ---------------------------------------------------------------------------
#define CW 32   // edges per chunk == lanes; E_CNT % CW == 0

__device__ __forceinline__ void async_stage_edges(const int* __restrict__ row,
                                                  const int* __restrict__ col,
                                                  const float* __restrict__ vals,
                                                  long base, int lane,
                                                  int4* lbuf /* shared, CW records */) {
  unsigned l0 = (unsigned)(unsigned long long)(const void*)(lbuf + lane);
  const int*   g0 = row  + base + lane;
  const int*   g1 = col  + base + lane;
  const float* g2 = vals + base + lane;
  asm volatile("global_load_async_to_lds_b32 %0, %1, off" :: "v"(l0),      "v"(g0) : "memory");
  asm volatile("global_load_async_to_lds_b32 %0, %1, off" :: "v"(l0 + 4u), "v"(g1) : "memory");
  asm volatile("global_load_async_to_lds_b32 %0, %1, off" :: "v"(l0 + 8u), "v"(g2) : "memory");
}

// ---------------------------------------------------------------------------
// SpMM via atomic scatter: y[row[e],:] += vals[e] * x[col[e],:]
// One wave (32 lanes) covers D=64 as float2 per lane. Edge stream is
// double-buffered through LDS with async-to-LDS loads; gather rows are
// prefetched per-lane ahead of their serial consumption.
// ---------------------------------------------------------------------------
__global__ void __launch_bounds__(256)
lightgcn_spmm(const int*   __restrict__ row,
              const int*   __restrict__ col,
              const float* __restrict__ vals,
              const float* __restrict__ x,
              float*       __restrict__ y) {
  __shared__ int4 s_edges[8][2][CW];      // [wave-in-block][buffer][edge record]
  const int lane = threadIdx.x & 31;
  const int wib  = __builtin_amdgcn_readfirstlane((int)(threadIdx.x >> 5));
  const int wave = blockIdx.x * (int)(blockDim.x >> 5) + wib;
  const int nwaves  = gridDim.x * (int)(blockDim.x >> 5);
  const int nchunks = E_CNT / CW;         // 300000, exact

  int chunk = wave;
  if (chunk >= nchunks) return;
  int cur = 0;
  async_stage_edges(row, col, vals, (long)chunk * CW, lane, &s_edges[wib][0][0]);

  for (; chunk < nchunks; chunk += nwaves) {
    const int nxt = chunk + nwaves;       // wave-uniform
    if (nxt < nchunks) {
      // Kick off next chunk's staging, then wait only for the current chunk
      // (async loads complete in order: <=3 outstanding -> current is ready).
      async_stage_edges(row, col, vals, (long)nxt * CW, lane,
                        &s_edges[wib][cur ^ 1][0]);
      asm volatile("s_wait_asynccnt 0x3" ::: "memory");
    } else {
      asm volatile("s_wait_asynccnt 0x0" ::: "memory");
    }

    const int4* cb = &s_edges[wib][cur][0];

    // Per-lane prefetch of this chunk's 32 gather rows (256B each) so the
    // serial j-loop's L2 latency overlaps with earlier edges' processing.
    {
      const int pc = ((const int*)cb)[lane * 4 + 1];   // lane's edge col (b32)
      const float* px = x + (long)pc * DIMS;
      __builtin_prefetch(px, 0, 3);
      __builtin_prefetch(px + 32, 0, 3);
    }

    for (int j = 0; j < CW; ++j) {
      const int4 t = cb[j];                 // one ds_load_b128 broadcast/edge
      const int   r = t.x;
      const int   c = t.y;
      const float v = __int_as_float(t.z);
      // Coalesced 256B gather: b64 per lane from the L2-resident x
      const float2 xv = ((const float2*)(x + (long)c * DIMS))[lane];
      float* yp = y + (long)r * DIMS + lane * 2;
      atomicAdd(yp + 0, v * xv.x);          // global_atomic_add_f32
      atomicAdd(yp + 1, v * xv.y);          // (no-return form)
    }
    cur ^= 1;
  }
}

// ---------------------------------------------------------------------------
// csum += y; emit csum*inv to the proper output segment(s);
// optionally zero the recycled ping-pong buffer for the next layer.
// ---------------------------------------------------------------------------
__global__ void lightgcn_accum_emit(float*       __restrict__ csum,
                                    const float* __restrict__ y,
                                    float*       __restrict__ zbuf,    // may be null
                                    float*       __restrict__ out_u,   // U*D segment
                                    float*       __restrict__ out_i,   // I*D segment
                                    float*       __restrict__ out_u2,  // may be null
                                    float*       __restrict__ out_i2,  // may be null
                                    float inv) {
  long i4 = (long)blockIdx.x * blockDim.x + threadIdx.x;
  const long n4 = ND / 4;
  if (i4 >= n4) return;
  float4 s = ((const float4*)csum)[i4];
  float4 t = ((const float4*)y)[i4];
  s.x += t.x; s.y += t.y; s.z += t.z; s.w += t.w;
  ((float4*)csum)[i4] = s;
  float4 m = make_float4(s.x * inv, s.y * inv, s.z * inv, s.w * inv);
  const long u4 = UD / 4;   // UD divisible by 4
  if (i4 < u4) {
    ((float4*)out_u)[i4] = m;
    if (out_u2) ((float4*)out_u2)[i4] = m;
  } else {
    ((float4*)out_i)[i4 - u4] = m;
    if (out_i2) ((float4*)out_i2)[i4 - u4] = m;
  }
  if (zbuf) ((float4*)zbuf)[i4] = make_float4(0.f, 0.f, 0.f, 0.f);
}

// ---------------------------------------------------------------------------
// Launch: e0 -> 3x (spmm + fused accumulate/emit), ping-pong buffers in ws.
// ---------------------------------------------------------------------------
extern "C" void kernel_launch(void* const* d_in, const int* in_sizes, int n_in,
                              void* d_out, int out_size, void* d_ws, size_t ws_size,
                              hipStream_t stream) {
  const float* ue   = (const float*)d_in[0];
  const float* ie   = (const float*)d_in[1];
  const int*   row  = (const int*)d_in[2];
  const int*   col  = (const int*)d_in[3];
  const float* vals = (const float*)d_in[4];
  float* out = (float*)d_out;

  // Workspace: 3 buffers of N*D floats (230.4 MB total)
  float* buf0 = (float*)d_ws;        // x   (holds e0 first)
  float* buf1 = buf0 + ND;           // y   (spmm accumulator, zeroed)
  float* buf2 = buf1 + ND;           // csum

  // Output layout: users | items | um0 um1 um2 | im0 im1 im2
  float* users = out;
  float* items = out + UD;
  float* um    = out + ND;            // 3 x UD
  float* im    = out + ND + 3 * UD;   // 3 x ID

  const int  tb = 256;
  const long n4 = ND / 4;
  const int  eb = (int)((n4 + tb - 1) / tb);
  const int  spmm_blocks = 4096;      // 32768 waves, grid-stride over edge chunks

  lightgcn_init<<<eb, tb, 0, stream>>>(ue, ie, buf0, buf2, buf1);

  // Layer 1: h1 = A*e0 ; csum = e0+h1 ; emit /2 ; zero buf0 for reuse as next y
  lightgcn_spmm<<<spmm_blocks, tb, 0, stream>>>(row, col, vals, buf0, buf1);
  lightgcn_accum_emit<<<eb, tb, 0, stream>>>(buf2, buf1, buf0,
                                             um + 0 * UD, im + 0 * ID,
                                             nullptr, nullptr, 0.5f);

  // Layer 2: h2 = A*h1 (x=buf1 -> y=buf0) ; csum += h2 ; emit /3 ; zero buf1
  lightgcn_spmm<<<spmm_blocks, tb, 0, stream>>>(row, col, vals, buf1, buf0);
  lightgcn_accum_emit<<<eb, tb, 0, stream>>>(buf2, buf0, buf1,
                                             um + 1 * UD, im + 1 * ID,
                                             nullptr, nullptr, 1.0f / 3.0f);

  // Layer 3: h3 = A*h2 (x=buf0 -> y=buf1) ; csum += h3 ; emit /4 to mean slots
  // AND to users/items (identical values).
  lightgcn_spmm<<<spmm_blocks, tb, 0, stream>>>(row, col, vals, buf0, buf1);
  lightgcn_accum_emit<<<eb, tb, 0, stream>>>(buf2, buf1, nullptr,
                                             um + 2 * UD, im + 2 * ID,
                                             users, items, 0.25f);
}